// S_LSTM_7395933684450
// MI455X (gfx1250) — compile-verified
//
#include <hip/hip_runtime.h>
#include <hip/hip_bf16.h>

typedef float v2f __attribute__((ext_vector_type(2)));
typedef float v4f __attribute__((ext_vector_type(4)));
typedef float v8f __attribute__((ext_vector_type(8)));

#define HSZ 16
#define NL  2
#define NB  512
#define NT  512
#define ND  128
#define NTILE (NB / HSZ)   // 32 batch tiles of 16

__device__ __forceinline__ v8f wmma_f32(v2f a, v2f b, v8f c) {
    // D = A(16x4,f32) * B(4x16,f32) + C(16x16,f32)
    return __builtin_amdgcn_wmma_f32_16x16x4_f32(false, a, false, b, (short)0, c, false, false);
}

__device__ __forceinline__ float sigmoidf_fast(float x) {
    return 1.0f / (1.0f + __expf(-x));
}
__device__ __forceinline__ float tanhf_fast(float x) {
    return 1.0f - 2.0f / (__expf(2.0f * x) + 1.0f);
}

// ---------------------------------------------------------------------------
// Kernel 1: repack Wx{i,f,c,o} [L,128,16] into A-fragment order:
//   Wpack[((((l*4+g)*32 + j)*2 + hi)*16 + m)*2 + c] = Wx_g[l][4j+2hi+c][m]
// ---------------------------------------------------------------------------
__global__ void pack_wx(const float* __restrict__ Wxi, const float* __restrict__ Wxf,
                        const float* __restrict__ Wxc, const float* __restrict__ Wxo,
                        float* __restrict__ Wpack) {
    int idx = blockIdx.x * 256 + threadIdx.x;   // 0 .. 16383
    int c  = idx & 1;
    int m  = (idx >> 1) & 15;
    int hi = (idx >> 5) & 1;
    int j  = (idx >> 6) & 31;
    int g  = (idx >> 11) & 3;
    int l  = (idx >> 13) & 1;
    const float* W = (g == 0) ? Wxi : (g == 1) ? Wxf : (g == 2) ? Wxc : Wxo;
    int k = 4 * j + 2 * hi + c;
    Wpack[idx] = W[(l * ND + k) * HSZ + m];
}

// ---------------------------------------------------------------------------
// Kernel 2: P(t,l,g)^T = Wx^T * x(t)^T + b   via fp32 WMMA, K=128 (32 x k4).
// One wave per (t, batch-tile). Output stored in WMMA D-fragment layout:
//   P[((t*8 + l*4+g)*32 + tile)*256 + lane*8 + r]
// ---------------------------------------------------------------------------
__global__ __launch_bounds__(32) void xproj(
        const float* __restrict__ x, const float* __restrict__ Wpack,
        const float* __restrict__ bi, const float* __restrict__ bf,
        const float* __restrict__ bc, const float* __restrict__ bo,
        float* __restrict__ P) {
    const int lane = threadIdx.x;
    const int n  = lane & 15;
    const int hi = lane >> 4;
    const int wid  = blockIdx.x;
    const int t    = wid >> 5;       // / NTILE
    const int tile = wid & 31;
    const float* xrow = x + (((long)(tile * 16 + n)) * NT + t) * ND;

    v8f acc[8] = {};
    #pragma unroll 4
    for (int j = 0; j < 32; ++j) {
        // x is single-use streaming data: non-temporal
        v2f bv = __builtin_nontemporal_load((const v2f*)(xrow + 4 * j + 2 * hi));
        #pragma unroll
        for (int lg = 0; lg < 8; ++lg) {
            v2f av = *(const v2f*)(Wpack + ((((lg * 32 + j) * 2 + hi) * 16 + n) << 1));
            acc[lg] = wmma_f32(av, bv, acc[lg]);
        }
    }

    const float* bptr[4] = {bi, bf, bc, bo};
    #pragma unroll
    for (int lg = 0; lg < 8; ++lg) {
        const int l = lg >> 2, g = lg & 3;
        const float* bb = bptr[g] + l * HSZ + 8 * hi;   // bias along m = 8*hi + r
        float* dst = P + (((long)t * 8 + lg) * NTILE + tile) * 256 + lane * 8;
        v4f lo, hi4;
        lo[0] = acc[lg][0] + bb[0]; lo[1] = acc[lg][1] + bb[1];
        lo[2] = acc[lg][2] + bb[2]; lo[3] = acc[lg][3] + bb[3];
        hi4[0] = acc[lg][4] + bb[4]; hi4[1] = acc[lg][5] + bb[5];
        hi4[2] = acc[lg][6] + bb[6]; hi4[3] = acc[lg][7] + bb[7];
        *(v4f*)dst       = lo;      // regular store: keep resident in L2 for lstm_rec
        *(v4f*)(dst + 4) = hi4;
    }
}

// ---------------------------------------------------------------------------
// P-fragment load for one timestep (8 (l,g) D-layout fragments, NT hint:
// read-once data, evict-first so it doesn't thrash the WGP cache).
// ---------------------------------------------------------------------------
__device__ __forceinline__ void load_pfrag(const float* __restrict__ P,
                                           int t, int tile, int lane, v8f pf[8]) {
    #pragma unroll
    for (int lg = 0; lg < 8; ++lg) {
        const float* pp = P + (((long)t * 8 + lg) * NTILE + tile) * 256 + lane * 8;
        v4f lo = __builtin_nontemporal_load((const v4f*)pp);
        v4f h4 = __builtin_nontemporal_load((const v4f*)(pp + 4));
        v8f a;
        a[0] = lo[0]; a[1] = lo[1]; a[2] = lo[2]; a[3] = lo[3];
        a[4] = h4[0]; a[5] = h4[1]; a[6] = h4[2]; a[7] = h4[3];
        pf[lg] = a;
    }
}

// ---------------------------------------------------------------------------
// One LSTM timestep for a 16-batch tile (transposed state, fp32 WMMA K=16).
// ---------------------------------------------------------------------------
__device__ __forceinline__ void lstm_step(
        int t, int tile, int lane, int n, int hi,
        const v8f pf[8],
        const v2f aWh0[4][4], const v2f aWh1[4][4], const v2f aWhs[4][4],
        v8f& c0, v8f& c1,
        float (*hs)[16][16],            // hs[2][16][16] in LDS
        float* __restrict__ out) {
    // ---- layer 0 ----
    v2f bf0[4];
    #pragma unroll
    for (int j = 0; j < 4; ++j)
        bf0[j] = *(const v2f*)&hs[0][n][4 * j + 2 * hi];

    v8f gate[4];
    #pragma unroll
    for (int g = 0; g < 4; ++g) {
        v8f acc = pf[g];                // l = 0
        #pragma unroll
        for (int j = 0; j < 4; ++j) acc = wmma_f32(aWh0[g][j], bf0[j], acc);
        gate[g] = acc;
    }
    v8f h0;
    #pragma unroll
    for (int r = 0; r < 8; ++r) {
        const float gi = sigmoidf_fast(gate[0][r]);
        const float gf = sigmoidf_fast(gate[1][r]);
        const float gc = tanhf_fast(gate[2][r]);
        const float go = sigmoidf_fast(gate[3][r]);
        const float cn = gf * c0[r] + gi * gc;
        c0[r] = cn;
        h0[r] = go * tanhf_fast(cn);
    }
    __syncthreads();
    {
        v4f s0, s1;
        s0[0] = h0[0]; s0[1] = h0[1]; s0[2] = h0[2]; s0[3] = h0[3];
        s1[0] = h0[4]; s1[1] = h0[5]; s1[2] = h0[6]; s1[3] = h0[7];
        *(v4f*)&hs[0][n][8 * hi]     = s0;   // b128 LDS stores
        *(v4f*)&hs[0][n][8 * hi + 4] = s1;
    }
    __syncthreads();

    // ---- layer 1 ----
    v2f bfA[4], bfB[4];
    #pragma unroll
    for (int j = 0; j < 4; ++j) {
        bfA[j] = *(const v2f*)&hs[0][n][4 * j + 2 * hi];  // fresh h0
        bfB[j] = *(const v2f*)&hs[1][n][4 * j + 2 * hi];  // h1 prev
    }
    #pragma unroll
    for (int g = 0; g < 4; ++g) {
        v8f acc = pf[4 + g];            // l = 1
        #pragma unroll
        for (int j = 0; j < 4; ++j) acc = wmma_f32(aWh1[g][j], bfB[j], acc);
        #pragma unroll
        for (int j = 0; j < 4; ++j) acc = wmma_f32(aWhs[g][j], bfA[j], acc);
        gate[g] = acc;
    }
    v8f h1;
    #pragma unroll
    for (int r = 0; r < 8; ++r) {
        const float gi = sigmoidf_fast(gate[0][r]);
        const float gf = sigmoidf_fast(gate[1][r]);
        const float gc = tanhf_fast(gate[2][r]);
        const float go = sigmoidf_fast(gate[3][r]);
        const float cn = gf * c1[r] + gi * gc;
        c1[r] = cn;
        h1[r] = go * tanhf_fast(cn);
    }
    __syncthreads();
    {
        v4f s0, s1;
        s0[0] = h1[0]; s0[1] = h1[1]; s0[2] = h1[2]; s0[3] = h1[3];
        s1[0] = h1[4]; s1[1] = h1[5]; s1[2] = h1[6]; s1[3] = h1[7];
        *(v4f*)&hs[1][n][8 * hi]     = s0;
        *(v4f*)&hs[1][n][8 * hi + 4] = s1;

        // out[b][t][h]: 8 contiguous floats per lane, write-only -> NT store
        float* op = out + (((long)(tile * 16 + n) * NT + t) * HSZ) + 8 * hi;
        __builtin_nontemporal_store(s0, (v4f*)op);
        __builtin_nontemporal_store(s1, (v4f*)(op + 4));
    }
    __syncthreads();
}

// ---------------------------------------------------------------------------
// Kernel 3: recurrent chain. One wave owns a 16-row batch tile for all T.
// Double-buffered P prefetch: step t+1's fragments are issued before step t's
// compute, hiding global/L2 latency under a full step of WMMA work.
// ---------------------------------------------------------------------------
__global__ __launch_bounds__(32) void lstm_rec(
        const float* __restrict__ P,
        const float* __restrict__ Whi, const float* __restrict__ Whf,
        const float* __restrict__ Whc, const float* __restrict__ Who,
        const float* __restrict__ Whsi, const float* __restrict__ Whsf,
        const float* __restrict__ Whsc, const float* __restrict__ Whso,
        float* __restrict__ out) {
    const int lane = threadIdx.x;
    const int n  = lane & 15;     // batch-in-tile (B/C/D col), also m for A frags
    const int hi = lane >> 4;
    const int tile = blockIdx.x;

    __shared__ float hs[2][16][16];   // [layer][n][m]
    #pragma unroll
    for (int l = 0; l < 2; ++l) {
        v4f z = {};
        *(v4f*)&hs[l][n][8 * hi]     = z;
        *(v4f*)&hs[l][n][8 * hi + 4] = z;
    }

    // Preload recurrent-weight A-fragments: A[m][k] = Wh[k][m] (Wh^T)
    const float* Whp[4]  = {Whi, Whf, Whc, Who};
    const float* Whsp[4] = {Whsi, Whsf, Whsc, Whso};
    v2f aWh0[4][4], aWh1[4][4], aWhs[4][4];
    #pragma unroll
    for (int g = 0; g < 4; ++g)
        #pragma unroll
        for (int j = 0; j < 4; ++j) {
            const int k0 = 4 * j + 2 * hi;
            v2f a;
            a[0] = Whp[g][(0 * HSZ + k0) * HSZ + n];
            a[1] = Whp[g][(0 * HSZ + k0 + 1) * HSZ + n];
            aWh0[g][j] = a;
            a[0] = Whp[g][(1 * HSZ + k0) * HSZ + n];
            a[1] = Whp[g][(1 * HSZ + k0 + 1) * HSZ + n];
            aWh1[g][j] = a;
            a[0] = Whsp[g][k0 * HSZ + n];
            a[1] = Whsp[g][(k0 + 1) * HSZ + n];
            aWhs[g][j] = a;
        }

    v8f c0 = {}, c1 = {};
    __syncthreads();

    v8f pA[8], pB[8];
    load_pfrag(P, 0, tile, lane, pA);
    for (int t = 0; t < NT; t += 2) {
        // prefetch t+1 while computing t
        load_pfrag(P, t + 1, tile, lane, pB);
        lstm_step(t, tile, lane, n, hi, pA, aWh0, aWh1, aWhs, c0, c1, hs, out);
        // prefetch t+2 while computing t+1 (clamp final redundant prefetch)
        const int tn = (t + 2 < NT) ? (t + 2) : (NT - 1);
        load_pfrag(P, tn, tile, lane, pA);
        lstm_step(t + 1, tile, lane, n, hi, pB, aWh0, aWh1, aWhs, c0, c1, hs, out);
    }
}

extern "C" void kernel_launch(void* const* d_in, const int* in_sizes, int n_in,
                              void* d_out, int out_size, void* d_ws, size_t ws_size,
                              hipStream_t stream) {
    const float* x    = (const float*)d_in[0];
    const float* Wxi  = (const float*)d_in[1];
    const float* Wxf  = (const float*)d_in[2];
    const float* Wxc  = (const float*)d_in[3];
    const float* Wxo  = (const float*)d_in[4];
    const float* Whi  = (const float*)d_in[5];
    const float* Whf  = (const float*)d_in[6];
    const float* Whc  = (const float*)d_in[7];
    const float* Who  = (const float*)d_in[8];
    const float* Whsi = (const float*)d_in[9];
    const float* Whsf = (const float*)d_in[10];
    const float* Whsc = (const float*)d_in[11];
    const float* Whso = (const float*)d_in[12];
    const float* bi   = (const float*)d_in[13];
    const float* bf   = (const float*)d_in[14];
    const float* bc   = (const float*)d_in[15];
    const float* bo   = (const float*)d_in[16];

    float* Wpack = (float*)d_ws;                 // 16384 floats = 64 KB
    float* P     = (float*)d_ws + 16384;         // T*L*4*B*H floats = 128 MB

    hipLaunchKernelGGL(pack_wx, dim3(64), dim3(256), 0, stream,
                       Wxi, Wxf, Wxc, Wxo, Wpack);
    hipLaunchKernelGGL(xproj, dim3(NT * NTILE), dim3(32), 0, stream,
                       x, Wpack, bi, bf, bc, bo, P);
    hipLaunchKernelGGL(lstm_rec, dim3(NTILE), dim3(32), 0, stream,
                       P, Whi, Whf, Whc, Who, Whsi, Whsf, Whsc, Whso,
                       (float*)d_out);
}